// RoIPooling2D_2585570312364
// MI455X (gfx1250) — compile-verified
//
#include <hip/hip_runtime.h>
#include <hip/hip_bf16.h>

// RoI crop_and_resize (bilinear): feature [1,50,50,1024] f32 -> out [512,14,14,1024] f32.
//
// Roofline: 411 MB stores -> ~17.6us floor @ 23.3 TB/s HBM; feature (10.2 MB) is
// L2-resident (192 MB L2). ~2 FLOP per output byte => pure data-movement problem; no
// matrix-op formulation applies (gather indices vary per site, so no shared WMMA
// operand). Optimization = minimal per-wave instruction overhead around b128 streams:
//  - kernel 1 precomputes interpolation geometry (exact IEEE math) into d_ws as
//    16-byte AoS entries -> the streaming kernel reads each site's geometry with a
//    single wide scalar load (merged s_load_b512 across the unrolled row).
//  - kernel 2 streams one (roi,py) row per 256-thread block: unrolled 14-px loop of
//    4x coalesced global_load_b128 + reference-ordered lerps + global_store_b128
//    th:TH_STORE_NT (write-once output doesn't displace the hot feature map in L2).

#define NROIS   512
#define PH      14
#define PW      14
#define FH      50
#define FW      50
#define NC      1024
#define THREADS 256   // NC/4 float4 lanes

typedef float v4f __attribute__((ext_vector_type(4)));
typedef int   v4i __attribute__((ext_vector_type(4)));

// Packed geometry entry: {elemOff0, elemOff1, weight(bitcast), pad} = 16 bytes.
// yGeom[n*PH+py] : {y0*FW*NC, y1*FW*NC, wy, 0}
// xGeom[n*PW+px] : {x0*NC,    x1*NC,    wx, 0}

// ---------------------------------------------------------------------------
// Kernel 1: geometry tables (7168 sites), exact reference IEEE math.
// ---------------------------------------------------------------------------
__global__ __launch_bounds__(256) void roi_geom_kernel(
    const float* __restrict__ rois,
    const int*   __restrict__ img_size,
    v4i*         __restrict__ yGeom,   // [NROIS*PH]
    v4i*         __restrict__ xGeom)   // [NROIS*PW]
{
    const int idx = blockIdx.x * blockDim.x + threadIdx.x;
    if (idx >= NROIS * PH) return;
    const int n = idx / PH;
    const int i = idx - n * PH;

    const float hF = (float)img_size[0];
    const float wF = (float)img_size[1];

    const float by1 = rois[n * 4 + 0] / hF;
    const float bx1 = rois[n * 4 + 1] / wF;
    const float by2 = rois[n * 4 + 2] / hF;
    const float bx2 = rois[n * 4 + 3] / wF;

    const float t  = (float)i / (float)(PH - 1);
    const float ys = (by1 + t * (by2 - by1)) * (float)(FH - 1);
    const float xs = (bx1 + t * (bx2 - bx1)) * (float)(FW - 1);

    const float y0f = floorf(ys);
    const float x0f = floorf(xs);

    const int y0 = (int)fminf(fmaxf(y0f,        0.0f), (float)(FH - 1));
    const int y1 = (int)fminf(fmaxf(y0f + 1.0f, 0.0f), (float)(FH - 1));
    const int x0 = (int)fminf(fmaxf(x0f,        0.0f), (float)(FW - 1));
    const int x1 = (int)fminf(fmaxf(x0f + 1.0f, 0.0f), (float)(FW - 1));

    v4i gy, gx;
    gy.x = y0 * FW * NC;
    gy.y = y1 * FW * NC;
    gy.z = __float_as_int(ys - y0f);
    gy.w = 0;
    gx.x = x0 * NC;
    gx.y = x1 * NC;
    gx.z = __float_as_int(xs - x0f);
    gx.w = 0;
    yGeom[idx] = gy;
    xGeom[idx] = gx;
}

// ---------------------------------------------------------------------------
// Kernel 2: streaming interpolation. One block per (roi, py) output row.
// Geometry reads are uniform wide scalar loads; per px: 4 coalesced b128 loads,
// reference-ordered lerps, 1 NT b128 store. Fully unrolled for pipelining.
// ---------------------------------------------------------------------------
__global__ __launch_bounds__(THREADS) void roi_row_kernel(
    const float* __restrict__ feature,
    const v4i*   __restrict__ yGeom,
    const v4i*   __restrict__ xGeom,
    float*       __restrict__ out)
{
    const int row = blockIdx.x;          // n*PH + py
    const int n   = row / PH;

    const v4i   gy  = yGeom[row];        // uniform -> s_load_b128
    const int   r0  = gy.x;
    const int   r1  = gy.y;
    const float wy  = __int_as_float(gy.z);
    const float owy = 1.0f - wy;

    const int t = threadIdx.x;           // channels 4t..4t+3
    float* __restrict__ outrow = out + (size_t)row * (size_t)(PW * NC);
    const v4i* __restrict__ xg = xGeom + n * PW;   // 14 contiguous 16B entries

#pragma unroll
    for (int px = 0; px < PW; ++px) {
        const v4i   gx  = xg[px];        // uniform; merges into wide s_loads
        const int   c0  = gx.x;
        const int   c1  = gx.y;
        const float wx  = __int_as_float(gx.z);
        const float owx = 1.0f - wx;

        const v4f v00 = *((const v4f*)(feature + (size_t)(r0 + c0)) + t);
        const v4f v01 = *((const v4f*)(feature + (size_t)(r0 + c1)) + t);
        const v4f v10 = *((const v4f*)(feature + (size_t)(r1 + c0)) + t);
        const v4f v11 = *((const v4f*)(feature + (size_t)(r1 + c1)) + t);

        // Exact reference lerp ordering: x-lerps, then y-lerp.
        const v4f top = v00 * owx + v01 * wx;
        const v4f bot = v10 * owx + v11 * wx;
        const v4f res = top * owy + bot * wy;

        __builtin_nontemporal_store(res, (v4f*)(outrow + (size_t)px * NC) + t);
    }
}

// ---------------------------------------------------------------------------
// Fallback (ws too small): single-site-per-block kernel, geometry inline.
// ---------------------------------------------------------------------------
__global__ __launch_bounds__(THREADS) void roi_site_kernel(
    const float* __restrict__ feature,
    const float* __restrict__ rois,
    const int*   __restrict__ img_size,
    float*       __restrict__ out)
{
    const int site = blockIdx.x;
    const int n    = site / (PH * PW);
    const int rem  = site - n * (PH * PW);
    const int py   = rem / PW;
    const int px   = rem - py * PW;

    const float hF = (float)img_size[0];
    const float wF = (float)img_size[1];

    const float by1 = rois[n * 4 + 0] / hF;
    const float bx1 = rois[n * 4 + 1] / wF;
    const float by2 = rois[n * 4 + 2] / hF;
    const float bx2 = rois[n * 4 + 3] / wF;

    const float ty = (float)py / (float)(PH - 1);
    const float tx = (float)px / (float)(PW - 1);
    const float ys = (by1 + ty * (by2 - by1)) * (float)(FH - 1);
    const float xs = (bx1 + tx * (bx2 - bx1)) * (float)(FW - 1);

    const float y0f = floorf(ys);
    const float x0f = floorf(xs);
    const float wy  = ys - y0f;
    const float wx  = xs - x0f;

    const int y0 = (int)fminf(fmaxf(y0f,        0.0f), (float)(FH - 1));
    const int y1 = (int)fminf(fmaxf(y0f + 1.0f, 0.0f), (float)(FH - 1));
    const int x0 = (int)fminf(fmaxf(x0f,        0.0f), (float)(FW - 1));
    const int x1 = (int)fminf(fmaxf(x0f + 1.0f, 0.0f), (float)(FW - 1));

    const int t = threadIdx.x;
    const v4f v00 = *((const v4f*)(feature + (size_t)((y0 * FW + x0) * NC)) + t);
    const v4f v01 = *((const v4f*)(feature + (size_t)((y0 * FW + x1) * NC)) + t);
    const v4f v10 = *((const v4f*)(feature + (size_t)((y1 * FW + x0) * NC)) + t);
    const v4f v11 = *((const v4f*)(feature + (size_t)((y1 * FW + x1) * NC)) + t);

    const float owx = 1.0f - wx;
    const float owy = 1.0f - wy;
    const v4f top = v00 * owx + v01 * wx;
    const v4f bot = v10 * owx + v11 * wx;
    const v4f res = top * owy + bot * wy;

    __builtin_nontemporal_store(res, (v4f*)(out + (size_t)site * NC) + t);
}

extern "C" void kernel_launch(void* const* d_in, const int* in_sizes, int n_in,
                              void* d_out, int out_size, void* d_ws, size_t ws_size,
                              hipStream_t stream) {
    const float* feature  = (const float*)d_in[0];   // [1,50,50,1024] f32
    const float* rois     = (const float*)d_in[1];   // [512,4] f32
    const int*   img_size = (const int*)d_in[2];     // [2] i32
    float*       out      = (float*)d_out;           // [512,14,14,1024] f32

    const int    nrows = NROIS * PH;                            // 7168
    const size_t need  = (size_t)nrows * 2 * sizeof(v4i);       // 224 KB of tables

    if (d_ws != nullptr && ws_size >= need) {
        v4i* yGeom = (v4i*)d_ws;
        v4i* xGeom = yGeom + nrows;

        hipLaunchKernelGGL(roi_geom_kernel, dim3((nrows + 255) / 256), dim3(256),
                           0, stream, rois, img_size, yGeom, xGeom);

        hipLaunchKernelGGL(roi_row_kernel, dim3(nrows), dim3(THREADS), 0, stream,
                           feature, yGeom, xGeom, out);
    } else {
        hipLaunchKernelGGL(roi_site_kernel, dim3(NROIS * PH * PW), dim3(THREADS),
                           0, stream, feature, rois, img_size, out);
    }
}